// DirectHODLRBlocks_18614388261384
// MI455X (gfx1250) — compile-verified
//
#include <hip/hip_runtime.h>
#include <hip/hip_bf16.h>
#include <stddef.h>

// ---------------------------------------------------------------------------
// HODLR matvec for MI455X (gfx1250), fp32 path via V_WMMA_F32_16X16X4_F32.
//   y = blockdiag(leaf) x + s^2 * sum_l U_l (swap_c (U_l^T x))
// All GEMMs are mapped to 16x16 f32 WMMA tiles (one matrix per wave32).
// ---------------------------------------------------------------------------

#define NTOT   131072      // LEAF * 2^DEPTH
#define BATCH  32
#define LEAF   32
#define DEPTH  12

typedef float v2f __attribute__((ext_vector_type(2)));
typedef float v8f __attribute__((ext_vector_type(8)));

// D = A(16x4) * B(4x16) + C(16x16), fp32, wave32.
// Assumed VGPR layouts (ISA 7.12.2):
//   A: v0 = A[m][k0] (lanes 0-15) / A[m][k0+2] (lanes 16-31), v1 = k0+1 / k0+3
//   B: mirrored over K, lane index = column n
//   C/D: vgpr j holds row (j + 8*(lane>>4)), column (lane&15)
__device__ __forceinline__ v8f wmma4_f32(v2f a, v2f b, v8f c) {
#if __has_builtin(__builtin_amdgcn_wmma_f32_16x16x4_f32)
    return __builtin_amdgcn_wmma_f32_16x16x4_f32(
        /*neg_a=*/false, a, /*neg_b=*/false, b,
        /*c_mod=*/(short)0, c, /*reuse_a=*/false, /*reuse_b=*/false);
#else
    // compile-keepalive fallback (flags a missing builtin via wmma==0 in histogram)
    c[0] += a[0] * b[0] + a[1] * b[1];
    return c;
#endif
}

// ---------------------------------------------------------------------------
// Kernel 1: leaf block-diagonal   y[n, blk*32+t] = sum_s L[blk][t][s] x[n, blk*32+s]
// One wave per leaf block; 2x2 C-tiles of 16x16, K = 32 (8 WMMA steps per tile).
// ---------------------------------------------------------------------------
__global__ __launch_bounds__(32)
void hodlr_leaf_kernel(const float* __restrict__ leaf,
                       const float* __restrict__ x,
                       float* __restrict__ y) {
    const int blk  = blockIdx.x;          // 0 .. 4095
    const int lane = threadIdx.x;
    const int half = lane >> 4;
    const int id   = lane & 15;
    const float* __restrict__ Lb = leaf + (size_t)blk * (LEAF * LEAF);
    const int col0 = blk * LEAF;

    v8f acc[2][2] = {};
    for (int k0 = 0; k0 < LEAF; k0 += 4) {
        const int ka = k0 + 2 * half;
        v2f a[2], b[2];
        // A[m][k] = L[t=m][s=k]
        a[0].x = Lb[(size_t)(id)      * LEAF + ka];
        a[0].y = Lb[(size_t)(id)      * LEAF + ka + 1];
        a[1].x = Lb[(size_t)(16 + id) * LEAF + ka];
        a[1].y = Lb[(size_t)(16 + id) * LEAF + ka + 1];
        // B[k][n] = x[n][col0 + k]
        b[0].x = x[(size_t)(id)      * NTOT + col0 + ka];
        b[0].y = x[(size_t)(id)      * NTOT + col0 + ka + 1];
        b[1].x = x[(size_t)(16 + id) * NTOT + col0 + ka];
        b[1].y = x[(size_t)(16 + id) * NTOT + col0 + ka + 1];
        acc[0][0] = wmma4_f32(a[0], b[0], acc[0][0]);
        acc[0][1] = wmma4_f32(a[0], b[1], acc[0][1]);
        acc[1][0] = wmma4_f32(a[1], b[0], acc[1][0]);
        acc[1][1] = wmma4_f32(a[1], b[1], acc[1][1]);
    }
    // D[m][n] -> y[n][col0 + m]   (initializes y; levels accumulate on top)
    for (int mt = 0; mt < 2; ++mt)
        for (int nt = 0; nt < 2; ++nt)
            for (int j = 0; j < 8; ++j) {
                const int m = mt * 16 + j + 8 * half;
                const int n = nt * 16 + id;
                y[(size_t)n * NTOT + col0 + m] = acc[mt][nt][j];
            }
}

// ---------------------------------------------------------------------------
// Kernel 2: projection   T[g][rr][n] = sum_s U[g][s][rr] * x[n][g*bsz + s]
// One wave per (g, r-tile, n-tile, K-chunk).  Split-K partials -> workspace,
// folded by a deterministic reduction kernel (fixed summation order).
// ---------------------------------------------------------------------------
__global__ __launch_bounds__(32)
void hodlr_proj_kernel(const float* __restrict__ u,
                       const float* __restrict__ x,
                       float* __restrict__ t_out,
                       float* __restrict__ partial,
                       int bsz, int r, int RT, int KSPLIT, int CK) {
    int idx = blockIdx.x;
    const int kc = idx % KSPLIT; idx /= KSPLIT;
    const int nt = idx & 1;      idx >>= 1;
    const int rt = idx % RT;
    const int g  = idx / RT;
    const int lane = threadIdx.x;
    const int half = lane >> 4;
    const int id   = lane & 15;

    const float* __restrict__ ub = u + (size_t)g * bsz * r;
    const float* __restrict__ xb = x + (size_t)(nt * 16 + id) * NTOT + (size_t)g * bsz;

    const int  rr0     = rt * 16 + id;        // A-row handled by this lane
    const bool arow_ok = (rr0 < r);           // pad tiles past rank with zeros
    const int  kbeg    = kc * CK;
    const int  kend    = kbeg + CK;

    v8f acc = {};
    for (int k0 = kbeg; k0 < kend; k0 += 4) {
        const int ka = k0 + 2 * half;
        __builtin_prefetch(&ub[(size_t)(ka + 16) * r], 0, 0);   // stream U
        v2f a, b;
        a.x = arow_ok ? ub[(size_t)ka * r + rr0]       : 0.0f;  // A[m][k] = U[s=k][rr=m]
        a.y = arow_ok ? ub[(size_t)(ka + 1) * r + rr0] : 0.0f;
        b.x = xb[ka];                                            // B[k][n] = x[n][..+k]
        b.y = xb[ka + 1];
        acc = wmma4_f32(a, b, acc);
    }

    if (KSPLIT == 1) {
        for (int j = 0; j < 8; ++j) {
            const int rr = rt * 16 + j + 8 * half;
            if (rr < r)
                t_out[((size_t)g * r + rr) * BATCH + nt * 16 + id] = acc[j];
        }
    } else {
        float* __restrict__ pb =
            partial + (size_t)((((size_t)g * RT + rt) * 2 + nt) * KSPLIT + kc) * 256;
        for (int j = 0; j < 8; ++j)
            pb[(j + 8 * half) * 16 + id] = acc[j];
    }
}

__global__ __launch_bounds__(256)
void hodlr_reduce_kernel(const float* __restrict__ partial,
                         float* __restrict__ t_out,
                         int r, int RT, int KSPLIT) {
    const int tile = blockIdx.x;          // (g*RT + rt)*2 + nt
    const int nt   = tile & 1;
    const int rt   = (tile >> 1) % RT;
    const int g    = (tile >> 1) / RT;
    const int e    = threadIdx.x;         // 256 tile elements: m = e>>4, n = e&15
    const int m    = e >> 4;
    const int n    = e & 15;

    const float* __restrict__ pb = partial + (size_t)tile * KSPLIT * 256 + e;
    float s = 0.0f;
    for (int kc = 0; kc < KSPLIT; ++kc)   // fixed order => deterministic
        s += pb[(size_t)kc * 256];

    const int rr = rt * 16 + m;
    if (rr < r)
        t_out[((size_t)g * r + rr) * BATCH + nt * 16 + n] = s;
}

// ---------------------------------------------------------------------------
// Kernel 3: expansion   y[n][pos] += s^2 * sum_rr U[pos][rr] * T[g^1][rr][n]
// One wave per 16(row) x 16(batch) output tile; K = r (always a multiple of 4).
// ---------------------------------------------------------------------------
__global__ __launch_bounds__(32)
void hodlr_expand_kernel(const float* __restrict__ u,
                         const float* __restrict__ t_in,
                         const float* __restrict__ scale_ptr,
                         float* __restrict__ y,
                         int r, int shift /* log2(bsz) */) {
    int idx = blockIdx.x;                 // (NTOT/16) * 2 tiles
    const int nt = idx & 1;
    const int st = idx >> 1;
    const int lane = threadIdx.x;
    const int half = lane >> 4;
    const int id   = lane & 15;

    const int row0 = st * 16;             // 16-row tile never straddles a block (bsz>=32)
    const int g    = row0 >> shift;
    const int gsw  = g ^ 1;               // child swap t[:, :, ::-1, :]

    const float* __restrict__ tb = t_in + (size_t)gsw * r * BATCH;
    const float* __restrict__ ua = u + (size_t)row0 * r;
    const float s  = scale_ptr[0];
    const float s2 = s * s;

    v8f acc = {};
    for (int k0 = 0; k0 < r; k0 += 4) {   // r in {20,32,52,64}: multiple of 4
        const int ka = k0 + 2 * half;
        __builtin_prefetch(&ua[(size_t)(id + 16) * r + ka], 0, 0);  // stream U
        v2f a, b;
        a.x = ua[(size_t)id * r + ka];                     // A[m][k] = U[row0+m][k]
        a.y = ua[(size_t)id * r + ka + 1];
        b.x = tb[(size_t)ka * BATCH + nt * 16 + id];       // B[k][n] = T[g^1][k][n]
        b.y = tb[(size_t)(ka + 1) * BATCH + nt * 16 + id];
        acc = wmma4_f32(a, b, acc);
    }
    for (int j = 0; j < 8; ++j) {
        const int m = j + 8 * half;
        float* yp = &y[(size_t)(nt * 16 + id) * NTOT + row0 + m];
        *yp += s2 * acc[j];               // unique element per lane: plain RMW
    }
}

// ---------------------------------------------------------------------------
// Host-side orchestration (all launches on `stream`, graph-capture safe).
// Workspace layout: T buffer at d_ws (<= 10.5 MB), split-K partials at
// d_ws + 12 MB (<= 1 MB).  Total <= ~13.1 MB.
// ---------------------------------------------------------------------------
extern "C" void kernel_launch(void* const* d_in, const int* in_sizes, int n_in,
                              void* d_out, int out_size, void* d_ws, size_t ws_size,
                              hipStream_t stream) {
    (void)in_sizes; (void)n_in; (void)out_size; (void)ws_size;

    const float* x     = (const float*)d_in[0];
    const float* leaf  = (const float*)d_in[1];
    const float* scale = (const float*)d_in[2];
    const float* u[DEPTH];
    for (int i = 0; i < DEPTH; ++i) u[i] = (const float*)d_in[3 + i];

    float* y     = (float*)d_out;
    float* t_buf = (float*)d_ws;                        // max 4096*20*32 floats
    float* part  = (float*)d_ws + (3u << 20);           // +12 MB

    static const int ranks[DEPTH] = {64, 64, 64, 64, 64, 64, 64, 64, 64, 52, 32, 20};

    // 1) leaf block-diagonal term (initializes y)
    hodlr_leaf_kernel<<<NTOT / LEAF, 32, 0, stream>>>(leaf, x, y);

    // 2) twelve low-rank levels, coarsest (bsz=65536) to finest (bsz=32)
    for (int lvl = 0; lvl < DEPTH; ++lvl) {
        const int bsz    = LEAF << (DEPTH - 1 - lvl);
        const int r      = ranks[lvl];
        const int G      = NTOT / bsz;            // number of child blocks (p*2)
        const int RT     = (r + 15) / 16;
        const int CK     = bsz < 1024 ? bsz : 1024;
        const int KSPLIT = bsz / CK;
        const int shift  = __builtin_ctz(bsz);

        hodlr_proj_kernel<<<dim3(G * RT * 2 * KSPLIT), 32, 0, stream>>>(
            u[lvl], x, t_buf, part, bsz, r, RT, KSPLIT, CK);

        if (KSPLIT > 1)
            hodlr_reduce_kernel<<<dim3(G * RT * 2), 256, 0, stream>>>(
                part, t_buf, r, RT, KSPLIT);

        hodlr_expand_kernel<<<dim3((NTOT / 16) * 2), 32, 0, stream>>>(
            u[lvl], t_buf, scale, y, r, shift);
    }
}